// PromptRoutingEmbedding_41515153883407
// MI455X (gfx1250) — compile-verified
//
#include <hip/hip_runtime.h>
#include <hip/hip_bf16.h>
#include <math.h>

#define B_ 256
#define S_ 512
#define D_ 1024
#define R_ 8
#define RP_ 16   // routes padded to 16 for the WMMA N dimension
#define V_ 16

typedef __attribute__((ext_vector_type(2))) float v2f;
typedef __attribute__((ext_vector_type(8))) float v8f;

// ---------------------------------------------------------------------------
// Kernel 1: masked mean pool over sequence (HBM-bound phase: 512 MB streamed).
// sentence[b][d] = (sum_s mask[b][s] * embeds[b][s][d]) / max(sum_s mask, 1)
// Grid: (D/512, B), 256 threads; each thread owns 2 consecutive d and streams
// the full S range with nontemporal float2 loads (touched once -> don't
// pollute caches). Mask row staged in LDS; every thread accumulates the
// count itself (512 LDS adds, negligible), so no reduction is needed.
// ---------------------------------------------------------------------------
__global__ __launch_bounds__(256) void pool_kernel(const float* __restrict__ embeds,
                                                   const int* __restrict__ attn,
                                                   float* __restrict__ sent) {
  __shared__ float smask[S_];
  const int b = blockIdx.y;
  const int tid = threadIdx.x;
  for (int i = tid; i < S_; i += 256) smask[i] = (float)attn[b * S_ + i];
  __syncthreads();

  const int d0 = blockIdx.x * 512 + tid * 2;
  const float* p = embeds + (size_t)b * S_ * D_ + d0;

  float acc0 = 0.f, acc1 = 0.f, cnt = 0.f;
#pragma unroll 4
  for (int s = 0; s < S_; ++s) {
    float m = smask[s];
    v2f v = __builtin_nontemporal_load((const v2f*)(p + (size_t)s * D_));
    cnt += m;
    acc0 += m * v.x;
    acc1 += m * v.y;
  }
  cnt = fmaxf(cnt, 1.0f);
  sent[b * D_ + d0]     = acc0 / cnt;
  sent[b * D_ + d0 + 1] = acc1 / cnt;
}

// ---------------------------------------------------------------------------
// Kernel 2a: zero-pad router weights [8,1024] -> [16,1024] so the WMMA inner
// loop has no per-lane predication (EXEC stays all-1s with zero SALU churn).
// ---------------------------------------------------------------------------
__global__ __launch_bounds__(256) void pad_w_kernel(const float* __restrict__ W,
                                                    float* __restrict__ Wp) {
  const int idx = blockIdx.x * 256 + threadIdx.x;   // over RP_*D_ = 16384
  const int r = idx / D_;
  const int d = idx % D_;
  Wp[idx] = (r < R_) ? W[r * D_ + d] : 0.0f;
}

// ---------------------------------------------------------------------------
// Kernel 2b: router logits via V_WMMA_F32_16X16X4_F32.
// logits[b][n] = sum_d sent[b][d] * Wp[n][d]   (n = 0..15, routes 8..15 = 0)
// One wave per 16-row batch tile (16 blocks x 32 threads). K = 1024, 4/step.
// Lane layouts per CDNA5 ISA 7.12.2:
//   A 16x4 f32 : lane L -> M = L%16; (VGPR j, half = L/16) -> K = 2*half + j
//   B 4x16 f32 : lane L -> N = L%16; (VGPR j, half)        -> K = 2*half + j
//   C/D 16x16  : VGPR i, lane L -> M = i + 8*(L/16), N = L%16
// ---------------------------------------------------------------------------
__global__ __launch_bounds__(32) void router_wmma_kernel(const float* __restrict__ sent,
                                                         const float* __restrict__ Wp,
                                                         float* __restrict__ logits) {
  const int btile = blockIdx.x;       // 0..15 (16 batch rows each)
  const int lane  = threadIdx.x;      // 0..31
  const int half  = lane >> 4;
  const int l16   = lane & 15;

  const float* arow = sent + (size_t)(btile * 16 + l16) * D_ + 2 * half;
  const float* brow = Wp   + (size_t)l16 * D_ + 2 * half;

  v8f c = {};
  for (int k0 = 0; k0 < D_; k0 += 4) {
    v2f a = *(const v2f*)(arow + k0);
    v2f b = *(const v2f*)(brow + k0);
    c = __builtin_amdgcn_wmma_f32_16x16x4_f32(
        /*neg_a=*/false, a, /*neg_b=*/false, b,
        /*c_mod=*/(short)0, c, /*reuse_a=*/false, /*reuse_b=*/false);
  }

#pragma unroll
  for (int i = 0; i < 8; ++i) {
    int m = i + 8 * half;                           // 0..15 within tile
    logits[(btile * 16 + m) * RP_ + l16] = c[i];    // padded [B,16] store
  }
}

// ---------------------------------------------------------------------------
// Kernel 3: softmax over 8 routes, top-2 (strict-> scans match
// jax.lax.top_k: descending, lowest index wins ties), then
// out[b][v][:] = p0*T[i0*16+v][:] + p1*T[i1*16+v][:].
// Grid: (V, B), 256 threads x float4 = 1024 d per block. Table is 512 KB
// total and L2-resident; this phase is write-bound (16 MB).
// ---------------------------------------------------------------------------
__global__ __launch_bounds__(256) void finalize_kernel(const float* __restrict__ logits,
                                                       const float* __restrict__ table,
                                                       float* __restrict__ out) {
  const int v = blockIdx.x;   // virtual token 0..15
  const int b = blockIdx.y;   // batch row

  float l[R_];
#pragma unroll
  for (int r = 0; r < R_; ++r) l[r] = logits[b * RP_ + r];

  float mx = l[0];
#pragma unroll
  for (int r = 1; r < R_; ++r) mx = fmaxf(mx, l[r]);

  float e[R_];
  float sum = 0.f;
#pragma unroll
  for (int r = 0; r < R_; ++r) { e[r] = expf(l[r] - mx); sum += e[r]; }

  int i0 = 0; float v0 = e[0];
#pragma unroll
  for (int r = 1; r < R_; ++r) { if (e[r] > v0) { v0 = e[r]; i0 = r; } }
  int i1 = (i0 == 0) ? 1 : 0; float v1 = e[i1];
#pragma unroll
  for (int r = 0; r < R_; ++r) {
    if (r != i0 && e[r] > v1) { v1 = e[r]; i1 = r; }
  }
  const float inv = 1.0f / sum;
  const float p0 = v0 * inv;
  const float p1 = v1 * inv;

  const float4* t0 = (const float4*)(table + (size_t)(i0 * V_ + v) * D_);
  const float4* t1 = (const float4*)(table + (size_t)(i1 * V_ + v) * D_);
  float4* o = (float4*)(out + ((size_t)b * V_ + v) * D_);

  const int t = threadIdx.x;           // 256 threads * 4 floats = 1024
  float4 a = t0[t];
  float4 bb = t1[t];
  float4 r;
  r.x = p0 * a.x + p1 * bb.x;
  r.y = p0 * a.y + p1 * bb.y;
  r.z = p0 * a.z + p1 * bb.z;
  r.w = p0 * a.w + p1 * bb.w;
  o[t] = r;
}

// ---------------------------------------------------------------------------
// Launch. Inputs (setup_inputs order):
//   0 indices(i64, unused) 1 input_ids(i64, unused) 2 inputs_embeds f32[B,S,D]
//   3 attention_mask i32[B,S] 4 router_w f32[R,D] 5 emb_table f32[128,D]
// Output: f32 [B, V, D].
// Workspace: sent (1 MB) + Wp (64 KB) + logits (16 KB).
// ---------------------------------------------------------------------------
extern "C" void kernel_launch(void* const* d_in, const int* in_sizes, int n_in,
                              void* d_out, int out_size, void* d_ws, size_t ws_size,
                              hipStream_t stream) {
  (void)in_sizes; (void)n_in; (void)out_size; (void)ws_size;

  const float* embeds  = (const float*)d_in[2];
  const int*   attn    = (const int*)d_in[3];
  const float* routerw = (const float*)d_in[4];
  const float* table   = (const float*)d_in[5];

  float* sent   = (float*)d_ws;                     // [B, D]    1 MB
  float* wpad   = sent + (size_t)B_ * D_;           // [16, D]   64 KB
  float* logits = wpad + (size_t)RP_ * D_;          // [B, 16]   16 KB
  float* outp   = (float*)d_out;

  pool_kernel<<<dim3(D_ / 512, B_), 256, 0, stream>>>(embeds, attn, sent);
  pad_w_kernel<<<dim3((RP_ * D_) / 256), 256, 0, stream>>>(routerw, wpad);
  router_wmma_kernel<<<dim3(B_ / 16), 32, 0, stream>>>(sent, wpad, logits);
  finalize_kernel<<<dim3(V_, B_), 256, 0, stream>>>(logits, table, outp);
}